// AttnDecoderRNN_14645838479807
// MI455X (gfx1250) — compile-verified
//
#include <hip/hip_runtime.h>
#include <hip/hip_bf16.h>
#include <math.h>

typedef __attribute__((ext_vector_type(2))) float v2f;
typedef __attribute__((ext_vector_type(4))) float v4f;
typedef __attribute__((ext_vector_type(8))) float v8f;

#define HD 1024
#define SEQ 2048
#define VOC 50257

#define WMMA_F32(a, b, c) \
  __builtin_amdgcn_wmma_f32_16x16x4_f32(false, (a), false, (b), (short)0, (c), false, false)

// ---------------- 1) rnn_input = concat(emb[word], last_context) ----------
__global__ void k_build_input(const int* __restrict__ word,
                              const float* __restrict__ emb,
                              const float* __restrict__ last_ctx,
                              float* __restrict__ x) {
  int i = blockIdx.x * blockDim.x + threadIdx.x;
  if (i < HD)            x[i] = emb[(long)word[0] * HD + i];
  else if (i < 2 * HD)   x[i] = last_ctx[i - HD];
}

// ---------------- generic wave-per-row GEMV: out[r]=bias[r]+W[r,:]·vec ----
// K must be a multiple of 128 (float4 x 32 lanes). Streams W at HBM rate.
__global__ void k_gemv(const float* __restrict__ W, const float* __restrict__ vec,
                       const float* __restrict__ bias, float* __restrict__ out,
                       int rows, int K) {
  int wave = (blockIdx.x * blockDim.x + threadIdx.x) >> 5;
  int lane = threadIdx.x & 31;
  if (wave >= rows) return;
  const v4f* W4 = (const v4f*)(W + (long)wave * K);
  const v4f* x4 = (const v4f*)vec;
  int n4 = K >> 2;
  float acc = 0.f;
#pragma unroll 4
  for (int i = lane; i < n4; i += 32) {
    __builtin_prefetch((const void*)(W4 + i + 64), 0, 1);  // ~2 iters ahead
    v4f w = W4[i];
    v4f v = x4[i];
    acc += w.x * v.x + w.y * v.y + w.z * v.z + w.w * v.w;
  }
  acc += __shfl_xor(acc, 16, 32);
  acc += __shfl_xor(acc,  8, 32);
  acc += __shfl_xor(acc,  4, 32);
  acc += __shfl_xor(acc,  2, 32);
  acc += __shfl_xor(acc,  1, 32);
  if (lane == 0) out[wave] = acc + bias[wave];
}

// ---------------- GRU elementwise combine ---------------------------------
__global__ void k_gru_combine(const float* __restrict__ gi, const float* __restrict__ gh,
                              const float* __restrict__ hprev,
                              float* __restrict__ h_ws, float* __restrict__ h_out) {
  int i = blockIdx.x * blockDim.x + threadIdx.x;
  if (i >= HD) return;
  float r = 1.f / (1.f + expf(-(gi[i] + gh[i])));
  float z = 1.f / (1.f + expf(-(gi[HD + i] + gh[HD + i])));
  float n = tanhf(gi[2 * HD + i] + r * gh[2 * HD + i]);
  float h = (1.f - z) * n + z * hprev[i];
  h_ws[i]  = h;
  h_out[i] = h;
}

// ---------------- energy GEMM (WMMA fp32) fused with scores = energy·h1 ---
// 1024 waves; wave owns a 32x64 tile: 2 A row-groups x 4 B column strips.
// Software-pipelined: step k+4 fragments loaded before step k's 8 WMMAs.
// C init: c0x = ba[j], c1x = 0 (bias only in the first 16 rows). After K, dot rows with h1 and
// emit partial row-scores (fixed-order, deterministic).
__global__ void k_energy_scores(const float* __restrict__ enc,
                                const float* __restrict__ Wa,
                                const float* __restrict__ ba,
                                const float* __restrict__ h1,
                                float* __restrict__ partial /* [SEQ][16] */) {
  int wave = (blockIdx.x * blockDim.x + threadIdx.x) >> 5;
  int lane = threadIdx.x & 31;
  int hf   = lane >> 4;       // K-pair / output-row-half selector
  int lm   = lane & 15;       // M (for A) or N (for B/C/D)
  int mt = wave >> 4;         // 0..63  -> s0 = 32*mt
  int ns = wave & 15;         // 0..15  -> j0 = 64*ns
  int s0 = mt * 32;
  int j0 = ns * 64;

  const float* A0r = enc + (long)(s0 +      lm) * HD + hf * 2;
  const float* A1r = enc + (long)(s0 + 16 + lm) * HD + hf * 2;
  const float* B0r = Wa + (long)(j0 +      lm) * HD + hf * 2;
  const float* B1r = Wa + (long)(j0 + 16 + lm) * HD + hf * 2;
  const float* B2r = Wa + (long)(j0 + 32 + lm) * HD + hf * 2;
  const float* B3r = Wa + (long)(j0 + 48 + lm) * HD + hf * 2;

  float bv0 = ba[j0 + lm], bv1 = ba[j0 + 16 + lm];
  float bv2 = ba[j0 + 32 + lm], bv3 = ba[j0 + 48 + lm];
  v8f c00 = {bv0, bv0, bv0, bv0, bv0, bv0, bv0, bv0};
  v8f c01 = {bv1, bv1, bv1, bv1, bv1, bv1, bv1, bv1};
  v8f c02 = {bv2, bv2, bv2, bv2, bv2, bv2, bv2, bv2};
  v8f c03 = {bv3, bv3, bv3, bv3, bv3, bv3, bv3, bv3};
  v8f c10 = {};
  v8f c11 = {};
  v8f c12 = {};
  v8f c13 = {};

  // prologue loads (k = 0)
  v2f a0 = *(const v2f*)(A0r);
  v2f a1 = *(const v2f*)(A1r);
  v2f b0 = *(const v2f*)(B0r);
  v2f b1 = *(const v2f*)(B1r);
  v2f b2 = *(const v2f*)(B2r);
  v2f b3 = *(const v2f*)(B3r);

#pragma unroll 4
  for (int kk = 4; kk < HD; kk += 4) {
    // issue next step's loads before consuming current fragments
    v2f a0n = *(const v2f*)(A0r + kk);
    v2f a1n = *(const v2f*)(A1r + kk);
    v2f b0n = *(const v2f*)(B0r + kk);
    v2f b1n = *(const v2f*)(B1r + kk);
    v2f b2n = *(const v2f*)(B2r + kk);
    v2f b3n = *(const v2f*)(B3r + kk);
    c00 = WMMA_F32(a0, b0, c00);
    c10 = WMMA_F32(a1, b0, c10);
    c01 = WMMA_F32(a0, b1, c01);
    c11 = WMMA_F32(a1, b1, c11);
    c02 = WMMA_F32(a0, b2, c02);
    c12 = WMMA_F32(a1, b2, c12);
    c03 = WMMA_F32(a0, b3, c03);
    c13 = WMMA_F32(a1, b3, c13);
    a0 = a0n; a1 = a1n; b0 = b0n; b1 = b1n; b2 = b2n; b3 = b3n;
  }
  // epilogue (last K-step)
  c00 = WMMA_F32(a0, b0, c00);
  c10 = WMMA_F32(a1, b0, c10);
  c01 = WMMA_F32(a0, b1, c01);
  c11 = WMMA_F32(a1, b1, c11);
  c02 = WMMA_F32(a0, b2, c02);
  c12 = WMMA_F32(a1, b2, c12);
  c03 = WMMA_F32(a0, b3, c03);
  c13 = WMMA_F32(a1, b3, c13);

  // scores fold: row m of D lives in component (m&7), half (m>>3) of lanes.
  float h0v = h1[j0 + lm],      h1v = h1[j0 + 16 + lm];
  float h2v = h1[j0 + 32 + lm], h3v = h1[j0 + 48 + lm];
#pragma unroll
  for (int v = 0; v < 8; ++v) {
    float p0 = c00[v] * h0v + c01[v] * h1v + c02[v] * h2v + c03[v] * h3v;
    float p1 = c10[v] * h0v + c11[v] * h1v + c12[v] * h2v + c13[v] * h3v;
    p0 += __shfl_xor(p0, 1, 32);
    p0 += __shfl_xor(p0, 2, 32);
    p0 += __shfl_xor(p0, 4, 32);
    p0 += __shfl_xor(p0, 8, 32);
    p1 += __shfl_xor(p1, 1, 32);
    p1 += __shfl_xor(p1, 2, 32);
    p1 += __shfl_xor(p1, 4, 32);
    p1 += __shfl_xor(p1, 8, 32);
    if (lm == 0) {
      partial[(long)(s0 +      v + hf * 8) * 16 + ns] = p0;
      partial[(long)(s0 + 16 + v + hf * 8) * 16 + ns] = p1;
    }
  }
}

// ---------------- softmax over 2048 scores (single block, 512 thr) --------
__global__ void k_softmax(const float* __restrict__ partial,
                          float* __restrict__ attn_ws, float* __restrict__ attn_out) {
  __shared__ float sc[SEQ];
  __shared__ float red[512];
  int tid = threadIdx.x;
  float lmax = -1e30f;
  for (int idx = tid; idx < SEQ; idx += 512) {
    const v4f* p4 = (const v4f*)(partial + (long)idx * 16);
    v4f s4 = p4[0];
    v4f t4 = p4[1];
    s4 += p4[2];
    t4 += p4[3];
    s4 += t4;
    float s = s4.x + s4.y + s4.z + s4.w;
    sc[idx] = s;
    lmax = fmaxf(lmax, s);
  }
  red[tid] = lmax; __syncthreads();
  for (int st = 256; st > 0; st >>= 1) {
    if (tid < st) red[tid] = fmaxf(red[tid], red[tid + st]);
    __syncthreads();
  }
  float gmax = red[0]; __syncthreads();
  float lsum = 0.f;
  for (int idx = tid; idx < SEQ; idx += 512) lsum += expf(sc[idx] - gmax);
  red[tid] = lsum; __syncthreads();
  for (int st = 256; st > 0; st >>= 1) {
    if (tid < st) red[tid] += red[tid + st];
    __syncthreads();
  }
  float inv = 1.f / red[0];
  for (int idx = tid; idx < SEQ; idx += 512) {
    float w = expf(sc[idx] - gmax) * inv;
    attn_ws[idx] = w;
    attn_out[idx] = w;
  }
}

// ---------------- context = attn @ enc (s-chunked partials) ---------------
__global__ void k_context_part(const float* __restrict__ attn, const float* __restrict__ enc,
                               float* __restrict__ ctxpart /* [16][HD] */) {
  int sb = blockIdx.x >> 2;                        // 0..15 s-chunk
  int j  = (blockIdx.x & 3) * 256 + threadIdx.x;   // 0..1023
  int s0 = sb * 128;
  float acc = 0.f;
#pragma unroll 4
  for (int s = s0; s < s0 + 128; ++s) acc += attn[s] * enc[(long)s * HD + j];
  ctxpart[(long)sb * HD + j] = acc;
}
__global__ void k_context_reduce(const float* __restrict__ ctxpart,
                                 float* __restrict__ ctx_ws, float* __restrict__ ctx_out) {
  int j = blockIdx.x * blockDim.x + threadIdx.x;   // 0..1023
  float acc = 0.f;
#pragma unroll
  for (int t = 0; t < 16; ++t) acc += ctxpart[(long)t * HD + j];
  ctx_ws[j] = acc;
  ctx_out[j] = acc;
}

// ---------------- log-softmax reductions ----------------------------------
__global__ void k_reduce_max(const float* __restrict__ x, int n, float* __restrict__ bout) {
  __shared__ float red[256];
  int tid = threadIdx.x;
  float m = -1e30f;
  for (int i = blockIdx.x * 256 + tid; i < n; i += gridDim.x * 256) m = fmaxf(m, x[i]);
  red[tid] = m; __syncthreads();
  for (int st = 128; st > 0; st >>= 1) {
    if (tid < st) red[tid] = fmaxf(red[tid], red[tid + st]);
    __syncthreads();
  }
  if (tid == 0) bout[blockIdx.x] = red[0];
}
__global__ void k_final_max(const float* __restrict__ bmax, float* __restrict__ gmax) {
  __shared__ float red[128];
  int tid = threadIdx.x;
  red[tid] = bmax[tid]; __syncthreads();
  for (int st = 64; st > 0; st >>= 1) {
    if (tid < st) red[tid] = fmaxf(red[tid], red[tid + st]);
    __syncthreads();
  }
  if (tid == 0) gmax[0] = red[0];
}
__global__ void k_reduce_sumexp(const float* __restrict__ x, int n,
                                const float* __restrict__ gmax, float* __restrict__ bout) {
  __shared__ float red[256];
  float g = gmax[0];
  int tid = threadIdx.x;
  float s = 0.f;
  for (int i = blockIdx.x * 256 + tid; i < n; i += gridDim.x * 256) s += expf(x[i] - g);
  red[tid] = s; __syncthreads();
  for (int st = 128; st > 0; st >>= 1) {
    if (tid < st) red[tid] += red[tid + st];
    __syncthreads();
  }
  if (tid == 0) bout[blockIdx.x] = red[0];
}
__global__ void k_final_lse(const float* __restrict__ bsum, const float* __restrict__ gmax,
                            float* __restrict__ lse) {
  __shared__ float red[128];
  int tid = threadIdx.x;
  red[tid] = bsum[tid]; __syncthreads();
  for (int st = 64; st > 0; st >>= 1) {
    if (tid < st) red[tid] += red[tid + st];
    __syncthreads();
  }
  if (tid == 0) lse[0] = gmax[0] + logf(red[0]);
}
__global__ void k_logsoftmax(const float* __restrict__ logits, const float* __restrict__ lse,
                             float* __restrict__ out, int n) {
  int i = blockIdx.x * blockDim.x + threadIdx.x;
  if (i < n) out[i] = logits[i] - lse[0];
}

extern "C" void kernel_launch(void* const* d_in, const int* in_sizes, int n_in,
                              void* d_out, int out_size, void* d_ws, size_t ws_size,
                              hipStream_t stream) {
  (void)in_sizes; (void)n_in; (void)out_size; (void)ws_size;
  const int*   word     = (const int*)  d_in[0];
  const float* last_ctx = (const float*)d_in[1];
  const float* last_hid = (const float*)d_in[2];  // (2,1,1024)
  const float* enc      = (const float*)d_in[3];  // (2048,1,1024)
  const float* emb      = (const float*)d_in[4];
  const float* W_ih0    = (const float*)d_in[5];
  const float* W_hh0    = (const float*)d_in[6];
  const float* b_ih0    = (const float*)d_in[7];
  const float* b_hh0    = (const float*)d_in[8];
  const float* W_ih1    = (const float*)d_in[9];
  const float* W_hh1    = (const float*)d_in[10];
  const float* b_ih1    = (const float*)d_in[11];
  const float* b_hh1    = (const float*)d_in[12];
  const float* Wa       = (const float*)d_in[13];
  const float* ba       = (const float*)d_in[14];
  const float* W_out    = (const float*)d_in[15];
  const float* b_out    = (const float*)d_in[16];

  float* out = (float*)d_out;
  float* o_output = out;             // 50257
  float* o_ctx    = out + 50257;     // 1024
  float* o_hid    = out + 51281;     // 2048 (h0 | h1)
  float* o_attn   = out + 53329;     // 2048

  float* ws = (float*)d_ws;          // all offsets 16B-aligned
  float* x       = ws;               // 2048
  float* gi0     = ws + 2048;        // 3072
  float* gh0     = ws + 5120;        // 3072
  float* gi1     = ws + 8192;        // 3072
  float* gh1     = ws + 11264;       // 3072
  float* y       = ws + 14336;       // 2048: [h1 | context]
  float* h0      = ws + 16384;       // 1024
  float* part    = ws + 17408;       // 2048*16
  float* attn    = ws + 50176;       // 2048
  float* logits  = ws + 52224;       // 50257
  float* bmax    = ws + 102496;      // 128
  float* bsum    = ws + 102624;      // 128
  float* gmax    = ws + 102752;      // 1 (padded)
  float* lse     = ws + 102756;      // 1 (padded)
  float* ctxpart = ws + 102784;      // 16*1024

  // 1) rnn_input
  k_build_input<<<8, 256, 0, stream>>>(word, emb, last_ctx, x);
  // 2) GRU0 gates + combine
  k_gemv<<<384, 256, 0, stream>>>(W_ih0, x, b_ih0, gi0, 3072, 2048);
  k_gemv<<<384, 256, 0, stream>>>(W_hh0, last_hid, b_hh0, gh0, 3072, 1024);
  k_gru_combine<<<4, 256, 0, stream>>>(gi0, gh0, last_hid, h0, o_hid);
  // 3) GRU1 gates + combine (h1 -> y[0:1024])
  k_gemv<<<384, 256, 0, stream>>>(W_ih1, h0, b_ih1, gi1, 3072, 1024);
  k_gemv<<<384, 256, 0, stream>>>(W_hh1, last_hid + HD, b_hh1, gh1, 3072, 1024);
  k_gru_combine<<<4, 256, 0, stream>>>(gi1, gh1, last_hid + HD, y, o_hid + HD);
  // 4) attention: WMMA energy GEMM fused with scores
  k_energy_scores<<<128, 256, 0, stream>>>(enc, Wa, ba, y, part);
  k_softmax<<<1, 512, 0, stream>>>(part, attn, o_attn);
  k_context_part<<<64, 256, 0, stream>>>(attn, enc, ctxpart);
  k_context_reduce<<<4, 256, 0, stream>>>(ctxpart, y + HD, o_ctx);
  // 5) output logits: streams 412MB of W_out -> bandwidth-bound
  k_gemv<<<6283, 256, 0, stream>>>(W_out, y, b_out, logits, VOC, 2048);
  // 6) log-softmax
  k_reduce_max<<<128, 256, 0, stream>>>(logits, VOC, bmax);
  k_final_max<<<1, 128, 0, stream>>>(bmax, gmax);
  k_reduce_sumexp<<<128, 256, 0, stream>>>(logits, VOC, gmax, bsum);
  k_final_lse<<<1, 128, 0, stream>>>(bsum, gmax, lse);
  k_logsoftmax<<<197, 256, 0, stream>>>(logits, lse, o_output, VOC);
}